// IntervalClusterTripletFT_48258252538457
// MI455X (gfx1250) — compile-verified
//
#include <hip/hip_runtime.h>
#include <math.h>
#include <stdint.h>

typedef __attribute__((ext_vector_type(2))) float v2f;
typedef __attribute__((ext_vector_type(8))) float v8f;

#define N_EMB   8192           // 512 clusters * 16
#define DIM     128
#define TILES   (N_EMB / 16)   // 512 column tiles == 512 clusters
#define LDS_STR 132            // 128 + 4 pad -> conflict-free ds_load_b64 gather
#define MARGIN_F 1.0f

// ---------------- kernel 1: squared norms ----------------
__global__ void sqnorm_kernel(const float* __restrict__ x, float* __restrict__ sq) {
    int row = blockIdx.x * blockDim.x + threadIdx.x;
    const float4* p = (const float4*)(x + (size_t)row * DIM);
    float acc = 0.f;
#pragma unroll
    for (int i = 0; i < DIM / 4; ++i) {
        float4 v = p[i];
        acc = fmaf(v.x, v.x, acc);
        acc = fmaf(v.y, v.y, acc);
        acc = fmaf(v.z, v.z, acc);
        acc = fmaf(v.w, v.w, acc);
    }
    sq[row] = acc;
}

// Async DMA: global -> LDS, bypassing VGPRs (tracked by ASYNCcnt).
// LDS dest = low 32 bits of the flat shared-memory address.
__device__ __forceinline__ void stage_tile_async(const float* __restrict__ x,
                                                 float* buf, int ct, int tid) {
#pragma unroll
    for (int i = 0; i < 4; ++i) {
        int s  = tid + i * 128;        // 512 float4 slots / 128 threads
        int r  = s >> 5;               // row within tile
        int c4 = s & 31;               // float4 column
        const float* g = x + (size_t)ct * 16 * DIM + r * DIM + c4 * 4;
        unsigned int l = (unsigned int)(uintptr_t)(buf + r * LDS_STR + c4 * 4);
        asm volatile("global_load_async_to_lds_b128 %0, %1, off"
                     :: "v"(l), "v"(g) : "memory");
    }
}

// ---------------- kernel 2: WMMA gram + hardest pos/neg + triplet loss ----------------
__global__ __launch_bounds__(128)
void triplet_kernel(const float* __restrict__ x, const float* __restrict__ sq,
                    float* __restrict__ out) {
    __shared__ float lds[2][16 * LDS_STR];

    const int tid  = threadIdx.x;
    const int wave = tid >> 5;       // 0..3
    const int lane = tid & 31;
    const int m    = lane & 15;      // column/row-in-tile index
    const int h    = lane >> 4;      // half select

    const int rt       = blockIdx.x * 4 + wave;  // row tile == cluster id
    const int row_base = rt * 16;

    // A fragment resident: 16 rows x 128 K of f32 across the wave (64 VGPRs/lane).
    // ISA layout A 16x4 f32: lane(m,h), VGPR v  <- A[m][2h + v].
    v2f a[32];
    {
        const float* arow = x + (size_t)(row_base + m) * DIM + 2 * h;
#pragma unroll
        for (int kb = 0; kb < 32; ++kb)
            a[kb] = *(const v2f*)(arow + 4 * kb);
    }

    float negmin[8], posmax[8];
#pragma unroll
    for (int v = 0; v < 8; ++v) { negmin[v] = __builtin_inff(); posmax[v] = -__builtin_inff(); }

    // Prologue: DMA tile 0 into buffer 0.
    stage_tile_async(x, lds[0], 0, tid);

    for (int ct = 0; ct < TILES; ++ct) {
        // My DMA for tile ct has landed; barrier makes every wave's staging visible.
        asm volatile("s_wait_asynccnt 0" ::: "memory");
        __syncthreads();

        // Kick off DMA for the next tile into the other buffer; it overlaps the
        // 32 WMMAs below. Safe: the barrier above also fenced everyone's reads
        // of that buffer from iteration ct-1.
        if (ct + 1 < TILES)
            stage_tile_async(x, lds[(ct + 1) & 1], ct + 1, tid);

        // 16x16 gram tile: 32 chained V_WMMA_F32_16X16X4_F32, 4-way K-split for XDL ILP.
        // B 4x16 f32 layout (transpose-symmetric to A): lane(m,h), VGPR v <- B[2h+v][m].
        v8f acc0 = {}, acc1 = {}, acc2 = {}, acc3 = {};
        const float* bcol = lds[ct & 1] + m * LDS_STR + 2 * h;
#pragma unroll
        for (int kb = 0; kb < 32; kb += 4) {
            v2f b0 = *(const v2f*)(bcol + 4 * (kb + 0));
            v2f b1 = *(const v2f*)(bcol + 4 * (kb + 1));
            v2f b2 = *(const v2f*)(bcol + 4 * (kb + 2));
            v2f b3 = *(const v2f*)(bcol + 4 * (kb + 3));
            acc0 = __builtin_amdgcn_wmma_f32_16x16x4_f32(false, a[kb + 0], false, b0, (short)0, acc0, false, false);
            acc1 = __builtin_amdgcn_wmma_f32_16x16x4_f32(false, a[kb + 1], false, b1, (short)0, acc1, false, false);
            acc2 = __builtin_amdgcn_wmma_f32_16x16x4_f32(false, a[kb + 2], false, b2, (short)0, acc2, false, false);
            acc3 = __builtin_amdgcn_wmma_f32_16x16x4_f32(false, a[kb + 3], false, b3, (short)0, acc3, false, false);
        }

        // Epilogue: t = sq_col - 2*gram  (sq_row added once at the end).
        // Tile rt==ct is the own-cluster (all-positive) tile; every other tile is all-negative.
        float scol = sq[ct * 16 + m];
        if (ct == rt) {
#pragma unroll
            for (int v = 0; v < 8; ++v) {
                float g = (acc0[v] + acc1[v]) + (acc2[v] + acc3[v]);
                float t = fmaf(-2.f, g, scol);
                posmax[v] = fmaxf(posmax[v], t);
            }
        } else {
#pragma unroll
            for (int v = 0; v < 8; ++v) {
                float g = (acc0[v] + acc1[v]) + (acc2[v] + acc3[v]);
                float t = fmaf(-2.f, g, scol);
                negmin[v] = fminf(negmin[v], t);
            }
        }
    }

    // Cross-lane reduction over the 16 columns each half of the wave saw.
#pragma unroll
    for (int off = 1; off < 16; off <<= 1) {
#pragma unroll
        for (int v = 0; v < 8; ++v) {
            negmin[v] = fminf(negmin[v], __shfl_xor(negmin[v], off, 16));
            posmax[v] = fmaxf(posmax[v], __shfl_xor(posmax[v], off, 16));
        }
    }

    // C-layout: VGPR v holds row v + 8*h. Lanes m==0 of each half hold reduced values.
    if (m == 0) {
#pragma unroll
        for (int v = 0; v < 8; ++v) {
            int   row = row_base + v + 8 * h;
            float sr  = sq[row];
            float dap = sqrtf(fmaxf(sr + posmax[v], 0.f));
            float dan = sqrtf(fmaxf(sr + negmin[v], 0.f));
            out[row]  = fmaxf(dap - dan + MARGIN_F, 0.f);
        }
    }
}

// ---------------- launch ----------------
extern "C" void kernel_launch(void* const* d_in, const int* in_sizes, int n_in,
                              void* d_out, int out_size, void* d_ws, size_t ws_size,
                              hipStream_t stream) {
    const float* x   = (const float*)d_in[0];   // [512,16,128] fp32, contiguous
    float*       out = (float*)d_out;           // [8192] fp32
    float*       sq  = (float*)d_ws;            // [8192] fp32 scratch (32 KB)

    sqnorm_kernel<<<N_EMB / 128, 128, 0, stream>>>(x, sq);
    triplet_kernel<<<TILES / 4, 128, 0, stream>>>(x, sq, out);
}